// MC3DAD_ONNX_48146583388946
// MI455X (gfx1250) — compile-verified
//
#include <hip/hip_runtime.h>

typedef __attribute__((ext_vector_type(2))) float v2f;
typedef __attribute__((ext_vector_type(4))) float v4f;
typedef __attribute__((ext_vector_type(8))) float v8f;

#define KNN 5
#define TILE 16

// One workgroup (256 threads = 8 waves) per (batch, 16-row tile).
// Each wave computes 16x16 Gram tiles with V_WMMA_F32_16X16X4_F32 (K=3 padded
// to 4), converts to squared distances in registers, stages them in LDS, and
// 256 threads (16 rows x 16 column-classes) maintain per-thread top-5 lists
// with branchless cndmask compare-swap chains. Final per-row merge ->
// covariance trace -> workspace.
__global__ void __launch_bounds__(256)
knn_trace_kernel(const float* __restrict__ pcd, float* __restrict__ trace, int N) {
  const int ntile = N / TILE;                 // 256
  const int b  = blockIdx.x / ntile;
  const int it = blockIdx.x % ntile;
  const int i0 = it * TILE;
  const float* P = pcd + (size_t)b * N * 3;

  const int tid  = threadIdx.x;
  const int w    = tid >> 5;                  // wave id 0..7
  const int lane = tid & 31;
  const int half = lane >> 4;                 // 0: lanes 0-15, 1: lanes 16-31
  const int lm   = lane & 15;

  // s_dist[(m*16+n)*8 + wb]: 8 candidates for (row m, col-class n) contiguous
  __shared__ __align__(16) float s_dist[8 * 256];
  __shared__ float s_d2[256 * KNN];
  __shared__ int   s_idx[256 * KNN];

  // ---- A fragment: rows i0..i0+15. 32-bit 16x4 layout:
  // lanes 0-15: v0=K0(x), v1=K1(y); lanes 16-31: v0=K2(z), v1=K3(0)
  const float* pr = P + (size_t)(i0 + lm) * 3;
  v2f afrag;
  afrag.x = half ? pr[2] : pr[0];
  afrag.y = half ? 0.0f  : pr[1];

  // Per-lane squared norms of the 8 rows this half of the wave covers
  // (D layout: VGPR r holds row r + 8*half).
  float rsq[8];
#pragma unroll
  for (int r = 0; r < 8; ++r) {
    const float* q = P + (size_t)(i0 + r + 8 * half) * 3;
    rsq[r] = q[0] * q[0] + q[1] * q[1] + q[2] * q[2];
  }

  // ---- top-5 scan state: thread t owns row m=t/16, column-class n=t%16
  const int m = tid >> 4;
  const int n = tid & 15;
  float bd[KNN]; int bj[KNN];
#pragma unroll
  for (int s = 0; s < KNN; ++s) { bd[s] = 3.402823466e+38f; bj[s] = 0x7fffffff; }

  const int nchunk = ntile / 8;               // 32: 8 column tiles per pass
  for (int c = 0; c < nchunk; ++c) {
    // B fragment for this wave's column tile (4x16, mirrors A layout)
    const int j0 = (c * 8 + w) * TILE;
    const float* pc = P + (size_t)(j0 + lm) * 3;
    const float qx = pc[0], qy = pc[1], qz = pc[2];
    v2f bfrag;
    bfrag.x = half ? qz : qx;
    bfrag.y = half ? 0.0f : qy;
    const float csq = qx * qx + qy * qy + qz * qz;

    v8f acc = {};
    acc = __builtin_amdgcn_wmma_f32_16x16x4_f32(
        /*neg_a=*/false, afrag, /*neg_b=*/false, bfrag,
        /*c_mod=*/(short)0, acc, /*reuse_a=*/false, /*reuse_b=*/false);

    // d^2 = |pi|^2 + |pj|^2 - 2 pi.pj   (no sqrt needed: same ordering)
#pragma unroll
    for (int r = 0; r < 8; ++r) {
      const float d2 = rsq[r] + csq - 2.0f * acc[r];
      s_dist[((r + 8 * half) * TILE + lm) * 8 + w] = d2;
    }
    __syncthreads();

    // Each thread folds its 8 contiguous candidates into its top-5.
    // Within a thread candidate indices are strictly increasing, so a strict
    // '<' compare reproduces top_k's stable (lowest-index-first) tie-breaks.
    {
      const v4f* p4 = (const v4f*)&s_dist[(m * TILE + n) * 8];
      const v4f d4a = p4[0];
      const v4f d4b = p4[1];
      float cand[8] = {d4a.x, d4a.y, d4a.z, d4a.w, d4b.x, d4b.y, d4b.z, d4b.w};
#pragma unroll
      for (int wb = 0; wb < 8; ++wb) {
        float d = cand[wb];
        int   j = (c * 8 + wb) * TILE + n;
        if (__any(d < bd[KNN - 1])) {       // wave-uniform skip, EXEC untouched
#pragma unroll
          for (int s = 0; s < KNN; ++s) {
            const bool  lt = d < bd[s];
            const float nd = lt ? bd[s] : d;
            const int   nj = lt ? bj[s] : j;
            bd[s] = lt ? d : bd[s];
            bj[s] = lt ? j : bj[s];
            d = nd; j = nj;
          }
        }
      }
    }
    __syncthreads();
  }

  // ---- dump per-thread lists, merge 16 lists per row
#pragma unroll
  for (int s = 0; s < KNN; ++s) {
    s_d2[tid * KNN + s]  = bd[s];
    s_idx[tid * KNN + s] = bj[s];
  }
  __syncthreads();

  if (tid < TILE) {
    const int row = tid;
    float fd[KNN]; int fj[KNN];
#pragma unroll
    for (int s = 0; s < KNN; ++s) { fd[s] = 3.402823466e+38f; fj[s] = 0x7fffffff; }
    for (int t = 0; t < TILE; ++t) {
      const int base = (row * TILE + t) * KNN;
#pragma unroll
      for (int s = 0; s < KNN; ++s) {
        float d = s_d2[base + s]; int j = s_idx[base + s];
#pragma unroll
        for (int u = 0; u < KNN; ++u) {
          // lexicographic (d, idx): exact lax.top_k tie semantics
          const bool  lt = (d < fd[u]) | ((d == fd[u]) & (j < fj[u]));
          const float nd = lt ? fd[u] : d;
          const int   nj = lt ? fj[u] : j;
          fd[u] = lt ? d : fd[u];
          fj[u] = lt ? j : fj[u];
          d = nd; j = nj;
        }
      }
    }

    // Gather 5 neighbors, centroid, covariance trace = sum ||p-c||^2 / (k-1)
    float nx[KNN], ny[KNN], nz[KNN];
    float cx = 0.0f, cy = 0.0f, cz = 0.0f;
#pragma unroll
    for (int s = 0; s < KNN; ++s) {
      const float* q = P + (size_t)fj[s] * 3;
      nx[s] = q[0]; ny[s] = q[1]; nz[s] = q[2];
      cx += nx[s]; cy += ny[s]; cz += nz[s];
    }
    const float inv_k = 1.0f / (float)KNN;
    cx *= inv_k; cy *= inv_k; cz *= inv_k;
    float tr = 0.0f;
#pragma unroll
    for (int s = 0; s < KNN; ++s) {
      const float dx = nx[s] - cx, dy = ny[s] - cy, dz = nz[s] - cz;
      tr += dx * dx + dy * dy + dz * dz;
    }
    tr *= 1.0f / (float)(KNN - 1);
    trace[(size_t)b * N + i0 + row] = tr;
  }
}

// Per-batch: curvature = trace / (sum(trace) + 1e-8)
__global__ void __launch_bounds__(256)
normalize_kernel(const float* __restrict__ trace, float* __restrict__ out, int N) {
  const int b = blockIdx.x;
  __shared__ float red[256];
  float s = 0.0f;
  for (int i = threadIdx.x; i < N; i += 256) s += trace[(size_t)b * N + i];
  red[threadIdx.x] = s;
  __syncthreads();
  for (int off = 128; off > 0; off >>= 1) {
    if (threadIdx.x < off) red[threadIdx.x] += red[threadIdx.x + off];
    __syncthreads();
  }
  const float denom = red[0] + 1e-8f;
  for (int i = threadIdx.x; i < N; i += 256)
    out[(size_t)b * N + i] = trace[(size_t)b * N + i] / denom;
}

extern "C" void kernel_launch(void* const* d_in, const int* in_sizes, int n_in,
                              void* d_out, int out_size, void* d_ws, size_t ws_size,
                              hipStream_t stream) {
  const float* pcd = (const float*)d_in[0];
  // d_in[1] is k==5 (compile-time KNN); top-k structure requires static k.
  float* trace = (float*)d_ws;                 // B*N floats (128 KB)
  float* out   = (float*)d_out;

  const int N = 4096;
  const int B = in_sizes[0] / (N * 3);

  dim3 grid1(B * (N / TILE));                  // 2048 workgroups
  knn_trace_kernel<<<grid1, 256, 0, stream>>>(pcd, trace, N);
  normalize_kernel<<<B, 256, 0, stream>>>(trace, out, N);
}